// GatEncoder_84361747628648
// MI455X (gfx1250) — compile-verified
//
#include <hip/hip_runtime.h>
#include <hip/hip_bf16.h>
#include <stdint.h>

// GAT-style encoder for MI455X (gfx1250, wave32, WMMA + async global->LDS).
// B=4, N=4096, IN=256, OUT=128, fp32 throughout (fp32 WMMA 16x16x4 keeps
// reference precision). Softmax over axis=1 factorizes:
//   coefs[i,j] = exp(lrelu(f1[i]+f2[j])) / denom[j],  denom depends only on j,
// so the [B,N,N] tensor is never materialized.
//
// B-operand data is stored K-pair-interleaved ("P layout"):
//   P[(k>>1)*2*COLS + col*2 + (k&1)] = M[k][col]
// so each lane's WMMA B fragment (two K-adjacent values at one column) is a
// single aligned ds_load_b64 into an even VGPR pair -- no packing movs.

#define BATCH 4
#define NN    4096
#define KIN   256
#define KOUT  128
#define NEG_SLOPE 0.01f
#define JCHUNK 64            // feats rows per async-staged LDS chunk (32KB)

typedef __attribute__((ext_vector_type(2))) float v2f;
typedef __attribute__((ext_vector_type(8))) float v8f;

__device__ __forceinline__ v8f wmma_f32_16x16x4(v2f a, v2f b, v8f c) {
    return __builtin_amdgcn_wmma_f32_16x16x4_f32(
        /*neg_a=*/false, a, /*neg_b=*/false, b,
        /*c_mod=*/(short)0, c, /*reuse_a=*/false, /*reuse_b=*/false);
}

// ---- gfx1250 async global->LDS helpers (ASYNCcnt-tracked, no VGPR bounce) ----
typedef __attribute__((address_space(3))) void lds_void_t;

__device__ __forceinline__ uint32_t lds_addr(void* p) {
    // generic -> LDS addrspace cast; AS3 pointer value IS the 32-bit LDS offset
    return (uint32_t)(uintptr_t)(lds_void_t*)p;
}
__device__ __forceinline__ void async_b128(uint32_t lds_off, const void* gaddr) {
    asm volatile("global_load_async_to_lds_b128 %0, %1, off"
                 :: "v"(lds_off), "v"(gaddr) : "memory");
}
__device__ __forceinline__ void wait_async_le(int n) {
    if (n == 0) asm volatile("s_wait_asynccnt 0x0" ::: "memory");
    else        asm volatile("s_wait_asynccnt 0x8" ::: "memory");
}

// ---------------------------------------------------------------------------
// Kernel 0: pre-pack W_fc [256,128] into K-pair-interleaved Wp.
// ---------------------------------------------------------------------------
__global__ __launch_bounds__(256) void wpack_kernel(
    const float* __restrict__ Wfc, float* __restrict__ Wp) {
    const int idx = blockIdx.x * 256 + threadIdx.x;   // [0, 32768)
    const int k = idx >> 7, o = idx & 127;
    Wp[(k >> 1) * (2 * KOUT) + o * 2 + (k & 1)] = Wfc[idx];
}

// ---------------------------------------------------------------------------
// Kernel 1: feats = x @ W_fc.  Wp (128KB, paired) async-staged into LDS once
// per block; each wave computes a 16(row) x 128(col) tile, K-loop step 4,
// 8 WMMA_F32_16X16X4 per step.  Output written in P layout (featsP).
// A-fragment layout (ISA 7.12.2): lane<16 -> row=lane, K={0,1};
// lane>=16 -> row=lane-16, K={2,3}.
// ---------------------------------------------------------------------------
__global__ __launch_bounds__(256) void feats_gemm_kernel(
    const float* __restrict__ x, const float* __restrict__ Wp,
    float* __restrict__ featsP) {
    __shared__ float s_W[KIN * KOUT];          // 128 KB, paired layout

    {   // async stage: 8192 B128 transfers, 32 per thread
        uint32_t wbase = lds_addr(s_W);
#pragma unroll
        for (int s = 0; s < 32; ++s) {
            int q = threadIdx.x + 256 * s;
            async_b128(wbase + q * 16, Wp + q * 4);
        }
        wait_async_le(0);
    }
    __syncthreads();

    const int lane = threadIdx.x & 31;
    const int wave = threadIdx.x >> 5;
    const int m0 = (blockIdx.x * 8 + wave) * 16;   // 1024 tiles of 16 rows
    const int l15 = lane & 15;
    const int half = lane >> 4;                    // 0: K={0,1}  1: K={2,3}
    const int kOff = half * 2;

    v8f zero = {};
    v8f acc[8];
#pragma unroll
    for (int t = 0; t < 8; ++t) acc[t] = zero;

    const float* xrow = x + (size_t)(m0 + l15) * KIN;

    for (int k = 0; k < KIN; k += 4) {
        v2f a = *(const v2f*)(xrow + k + kOff);            // global_load_b64
        const float* w2 = s_W + ((k + kOff) >> 1) * (2 * KOUT);
#pragma unroll
        for (int t = 0; t < 8; ++t) {
            v2f b = *(const v2f*)(w2 + (t * 16 + l15) * 2); // ds_load_b64
            acc[t] = wmma_f32_16x16x4(a, b, acc[t]);
        }
    }

    // C/D layout: VGPR r, lanes0-15 -> (M=r, N=lane); lanes16-31 -> (M=r+8).
    // Write P layout: row = m0 + r + 8*half (parity = r&1).
#pragma unroll
    for (int t = 0; t < 8; ++t)
#pragma unroll
        for (int r = 0; r < 8; ++r) {
            const int row = m0 + r + 8 * half;
            featsP[(size_t)(row >> 1) * (2 * KOUT) + (t * 16 + l15) * 2 + (r & 1)] =
                acc[t][r];
        }
}

// ---------------------------------------------------------------------------
// Kernel 2: f1 = feats . w_l + b_l ; f2 = feats . w_r + b_r
// Consumes P layout: one thread per ROW PAIR; float4 covers 2 rows x 2 cols.
// ---------------------------------------------------------------------------
__global__ __launch_bounds__(256) void attn_proj_kernel(
    const float* __restrict__ featsP,
    const float* __restrict__ w_l, const float* __restrict__ b_l,
    const float* __restrict__ w_r, const float* __restrict__ b_r,
    float* __restrict__ f1, float* __restrict__ f2) {
    const int p = blockIdx.x * blockDim.x + threadIdx.x;  // row pair [0, 8192)
    const float4* fp = (const float4*)(featsP + (size_t)p * (2 * KOUT));
    const float2* wl2 = (const float2*)w_l;
    const float2* wr2 = (const float2*)w_r;
    float s1a = 0.f, s1b = 0.f, s2a = 0.f, s2b = 0.f;
#pragma unroll 4
    for (int q = 0; q < KOUT / 2; ++q) {
        float4 v = fp[q];          // {f[2p][2q], f[2p+1][2q], f[2p][2q+1], f[2p+1][2q+1]}
        float2 a = wl2[q], c = wr2[q];
        s1a += v.x * a.x + v.z * a.y;
        s1b += v.y * a.x + v.w * a.y;
        s2a += v.x * c.x + v.z * c.y;
        s2b += v.y * c.x + v.w * c.y;
    }
    f1[2 * p]     = s1a + b_l[0];
    f1[2 * p + 1] = s1b + b_l[0];
    f2[2 * p]     = s2a + b_r[0];
    f2[2 * p + 1] = s2b + b_r[0];
}

// ---------------------------------------------------------------------------
// Kernel 3: rdenom[b,j] = 1 / sum_i exp(lrelu(f1[b,i] + f2[b,j]))
// ---------------------------------------------------------------------------
__global__ __launch_bounds__(256) void denom_kernel(
    const float* __restrict__ f1, const float* __restrict__ f2,
    float* __restrict__ rdenom) {
    __shared__ float s_f1[NN];
    const int b = blockIdx.x >> 4;                 // 16 blocks per batch
    const int j = ((blockIdx.x & 15) << 8) + threadIdx.x;
    const float* f1b = f1 + b * NN;
    for (int i = threadIdx.x; i < NN; i += 256) s_f1[i] = f1b[i];
    __syncthreads();

    const float f2v = f2[b * NN + j];
    float sum = 0.f;
#pragma unroll 4
    for (int i = 0; i < NN; ++i) {
        float d = s_f1[i] + f2v;
        float s = d >= 0.f ? d : NEG_SLOPE * d;
        sum += __expf(s);
    }
    rdenom[b * NN + j] = 1.0f / sum;
}

// ---------------------------------------------------------------------------
// Kernel 4: out[b,i,:] = sum_j exp(lrelu(f1[i]+f2[j])) * rdenom[j] * feats[b,j,:]
// Flash-style: the 16x4 A-tile of coefs is computed in registers per WMMA step.
// featsP chunks (already pair-interleaved) are async-staged global->LDS,
// double-buffered, shared by all 8 waves; B fragment = one ds_load_b64.
// (f2,rdenom) packed as float2 in LDS.
// ---------------------------------------------------------------------------
__global__ __launch_bounds__(256) void out_gemm_kernel(
    const float* __restrict__ featsP, const float* __restrict__ f1,
    const float* __restrict__ f2, const float* __restrict__ rdenom,
    float* __restrict__ out) {
    __shared__ float2 s_fr[NN];                 // (f2, rdenom) pairs, 32 KB
    __shared__ float  s_ft[2][JCHUNK * KOUT];   // double buffer, 2 x 32 KB (P layout)

    const int b = blockIdx.x >> 5;              // 32 blocks per batch
    const int group = blockIdx.x & 31;          // 8 i-tiles per block
    const float* fb = featsP + (size_t)b * NN * KOUT;

    // issue chunk 0 async loads (8 x b128 per thread)
    {
        uint32_t base = lds_addr(&s_ft[0][0]);
#pragma unroll
        for (int s = 0; s < 8; ++s) {
            int q = threadIdx.x + 256 * s;
            async_b128(base + q * 16, fb + q * 4);
        }
    }
    // stage (f2, rdenom) pairs
    {
        const float* f2b = f2 + b * NN;
        const float* rdb = rdenom + b * NN;
        for (int i = threadIdx.x; i < NN; i += 256)
            s_fr[i] = make_float2(f2b[i], rdb[i]);
    }

    const int lane = threadIdx.x & 31;
    const int wave = threadIdx.x >> 5;
    const int m0 = (group * 8 + wave) * 16;     // i-tile base row
    const int l15 = lane & 15;
    const int half = lane >> 4;
    const int kOff = half * 2;

    const float f1m = f1[b * NN + m0 + l15];    // per-lane constant row value

    v8f zero = {};
    v8f acc[8];
#pragma unroll
    for (int t = 0; t < 8; ++t) acc[t] = zero;

    const int NCH = NN / JCHUNK;                // 64 chunks
    for (int c = 0; c < NCH; ++c) {
        const int buf = c & 1;
        if (c + 1 < NCH) {                      // prefetch next chunk into buf^1
            uint32_t base = lds_addr(&s_ft[buf ^ 1][0]);
            const float* g = fb + (size_t)(c + 1) * JCHUNK * KOUT;
#pragma unroll
            for (int s = 0; s < 8; ++s) {
                int q = threadIdx.x + 256 * s;
                async_b128(base + q * 16, g + q * 4);
            }
            wait_async_le(8);                   // chunk c landed, c+1 in flight
        } else {
            wait_async_le(0);
        }
        __syncthreads();                        // all waves see chunk c in LDS

        const float* fbuf = s_ft[buf];
        const int jg0 = c * JCHUNK;
        for (int jr = 0; jr < JCHUNK; jr += 4) {
            const int ja = jg0 + jr + kOff;
            float2 p0 = s_fr[ja];
            float2 p1 = s_fr[ja + 1];
            float d0 = f1m + p0.x;
            float d1 = f1m + p1.x;
            d0 = d0 >= 0.f ? d0 : NEG_SLOPE * d0;
            d1 = d1 >= 0.f ? d1 : NEG_SLOPE * d1;
            v2f a;
            a.x = __expf(d0) * p0.y;
            a.y = __expf(d1) * p1.y;

            // P layout: pair index within chunk = jr/2 + half
            const float* r2 = fbuf + ((jr >> 1) + half) * (2 * KOUT);
#pragma unroll
            for (int t = 0; t < 8; ++t) {
                v2f bb = *(const v2f*)(r2 + (t * 16 + l15) * 2);  // ds_load_b64
                acc[t] = wmma_f32_16x16x4(a, bb, acc[t]);
            }
        }
        __syncthreads();                        // done reading buf before overwrite
    }

    // out is row-major [b, i, o]
    float* ob = out + (size_t)b * NN * KOUT;
#pragma unroll
    for (int t = 0; t < 8; ++t)
#pragma unroll
        for (int r = 0; r < 8; ++r)
            ob[(size_t)(m0 + r + 8 * half) * KOUT + t * 16 + l15] = acc[t][r];
}

// ---------------------------------------------------------------------------
extern "C" void kernel_launch(void* const* d_in, const int* in_sizes, int n_in,
                              void* d_out, int out_size, void* d_ws, size_t ws_size,
                              hipStream_t stream) {
    const float* x   = (const float*)d_in[0];   // [4,4096,256]
    const float* Wfc = (const float*)d_in[1];   // [256,128]
    const float* w_l = (const float*)d_in[2];   // [128]
    const float* b_l = (const float*)d_in[3];   // scalar
    const float* w_r = (const float*)d_in[4];   // [128]
    const float* b_r = (const float*)d_in[5];   // scalar
    float* out = (float*)d_out;                 // [4,4096,128]

    // workspace layout (~8.4 MB)
    float* featsP = (float*)d_ws;                          // 16384*128 (P layout)
    float* f1     = featsP + (size_t)BATCH * NN * KOUT;    // 16384
    float* f2     = f1 + BATCH * NN;                       // 16384
    float* rdenom = f2 + BATCH * NN;                       // 16384
    float* Wp     = rdenom + BATCH * NN;                   // 32768 (P layout)

    wpack_kernel<<<dim3(128), dim3(256), 0, stream>>>(Wfc, Wp);
    feats_gemm_kernel<<<dim3(128), dim3(256), 0, stream>>>(x, Wp, featsP);
    attn_proj_kernel<<<dim3(32), dim3(256), 0, stream>>>(featsP, w_l, b_l, w_r, b_r, f1, f2);
    denom_kernel<<<dim3(64), dim3(256), 0, stream>>>(f1, f2, rdenom);
    out_gemm_kernel<<<dim3(128), dim3(256), 0, stream>>>(featsP, f1, f2, rdenom, out);
}